// RouterModel_48644799595099
// MI455X (gfx1250) — compile-verified
//
#include <hip/hip_runtime.h>

typedef float v2f __attribute__((ext_vector_type(2)));
typedef float v8f __attribute__((ext_vector_type(8)));

#define N_TOK 32768
#define DIM   1024
#define ND    ((size_t)N_TOK * (size_t)DIM)

// Fused router: gate GEMM via V_WMMA_F32_16X16X4_F32, softmax-of-2/top-1,
// then one streaming pass writing x0, x1, xc.
__global__ __launch_bounds__(256) void router_fused_kernel(
    const float* __restrict__ x,   // [N_TOK, DIM]
    const float* __restrict__ W,   // [DIM, 2]
    float* __restrict__ out)       // [3, N_TOK, DIM] (x0 | x1 | xc)
{
    // Stage W transposed into LDS, padded to 4 rows (rows 2,3 = 0) so every
    // lane of the B-fragment reads valid LDS (keeps EXEC uniform at WMMA).
    __shared__ __align__(16) float Wt[4 * 1024];
    for (int i = threadIdx.x; i < 4 * 1024; i += 256) {
        int n = i >> 10;          // transposed row (= W column)
        int k = i & 1023;
        Wt[i] = (n < 2) ? W[k * 2 + n] : 0.0f;
    }
    __syncthreads();

    const int lane    = threadIdx.x & 31;
    const int wave    = threadIdx.x >> 5;
    const int m       = lane & 15;     // row (A) / column (B) within tile
    const int half    = lane >> 4;     // K-pair selector per ISA layout
    const int rowBase = blockIdx.x * 128 + wave * 16;

    const float* xrow = x + (size_t)(rowBase + m) * DIM;   // A-fragment row
    const int    bn   = (m < 2) ? m : 2;                   // zero row for n>=2
    const float* wrow = Wt + bn * 1024;                    // B-fragment column

    // K loop: 4 independent accumulators to avoid WMMA->WMMA RAW stalls.
    v8f acc0 = {}, acc1 = {}, acc2 = {}, acc3 = {};
    for (int k0 = 0; k0 < DIM; k0 += 16) {
        const int kk = k0 + 2 * half;
        v2f a0 = *(const v2f*)(xrow + kk);
        v2f b0 = *(const v2f*)(wrow + kk);
        v2f a1 = *(const v2f*)(xrow + kk + 4);
        v2f b1 = *(const v2f*)(wrow + kk + 4);
        v2f a2 = *(const v2f*)(xrow + kk + 8);
        v2f b2 = *(const v2f*)(wrow + kk + 8);
        v2f a3 = *(const v2f*)(xrow + kk + 12);
        v2f b3 = *(const v2f*)(wrow + kk + 12);
        acc0 = __builtin_amdgcn_wmma_f32_16x16x4_f32(false, a0, false, b0,
                                                     (short)0, acc0, false, false);
        acc1 = __builtin_amdgcn_wmma_f32_16x16x4_f32(false, a1, false, b1,
                                                     (short)0, acc1, false, false);
        acc2 = __builtin_amdgcn_wmma_f32_16x16x4_f32(false, a2, false, b2,
                                                     (short)0, acc2, false, false);
        acc3 = __builtin_amdgcn_wmma_f32_16x16x4_f32(false, a3, false, b3,
                                                     (short)0, acc3, false, false);
    }
    v8f c = (acc0 + acc1) + (acc2 + acc3);

    float* o0 = out;            // x0
    float* o1 = out + ND;       // x1
    float* oc = out + 2 * ND;   // xc

    // C/D layout: row M in VGPR M%8; N=0 -> lane 0 (M<8) / 16 (M>=8); N=1 -> lane 1/17.
    #pragma unroll
    for (int r = 0; r < 16; ++r) {
        const int   srcBase = (r < 8) ? 0 : 16;
        const float cm = c[r & 7];
        const float d0 = __shfl(cm, srcBase + 0, 32);
        const float d1 = __shfl(cm, srcBase + 1, 32);

        // argmax with first-index tie-break (matches jnp.argmax)
        const bool  p0   = (d0 >= d1);
        const float diff = p0 ? (d1 - d0) : (d0 - d1);      // = min - max <= 0
        const float g    = 1.0f / (1.0f + __expf(diff));    // softmax winner score
        const float g0   = p0 ? g : 0.0f;
        const float g1   = p0 ? 0.0f : g;

        const size_t off = (size_t)(rowBase + r) * DIM;
        const float4* xin = (const float4*)(x + off);       // warm in WGP$/L2
        float4* p00 = (float4*)(o0 + off);
        float4* p01 = (float4*)(o1 + off);
        float4* p0c = (float4*)(oc + off);

        #pragma unroll
        for (int it = 0; it < 8; ++it) {
            const int idx = it * 32 + lane;                 // coalesced 512B/wave
            const float4 v = xin[idx];
            p00[idx] = make_float4(v.x * g0, v.y * g0, v.z * g0, v.w * g0);
            p01[idx] = make_float4(v.x * g1, v.y * g1, v.z * g1, v.w * g1);
            p0c[idx] = make_float4(v.x * g,  v.y * g,  v.z * g,  v.w * g);
        }
    }
}

extern "C" void kernel_launch(void* const* d_in, const int* in_sizes, int n_in,
                              void* d_out, int out_size, void* d_ws, size_t ws_size,
                              hipStream_t stream) {
    const float* x = (const float*)d_in[0];   // [32768, 1024] f32
    const float* W = (const float*)d_in[1];   // [1024, 2] f32
    float* out = (float*)d_out;               // 3 * 32768 * 1024 f32

    dim3 grid(N_TOK / 128);   // 256 blocks, each = 8 waves x 16 rows
    dim3 block(256);
    router_fused_kernel<<<grid, block, 0, stream>>>(x, W, out);
}